// SemanticAwarePartialAttention_60292750901349
// MI455X (gfx1250) — compile-verified
//
#include <hip/hip_runtime.h>

typedef __attribute__((ext_vector_type(2))) float v2f;
typedef __attribute__((ext_vector_type(8))) float v8f;

#define BATCH 16
#define CH    32
#define NPOS  4096     // 64*64
#define KSEL  2048     // 0.5 * NPOS
#define EPSN  1e-12f

// Workspace layout (floats)
#define OFF_INV   0                      // BATCH*NPOS       inv norms
#define OFF_PART  (BATCH*NPOS)           // BATCH*16*CH      partial channel sums
#define OFF_S     (OFF_PART + BATCH*16*CH) // BATCH*CH       channel-mean (unscaled)
#define OFF_SCORE (OFF_S + BATCH*CH)     // BATCH*NPOS       scores
#define OFF_MASK  (OFF_SCORE + BATCH*NPOS) // BATCH*NPOS     0/1 mask

__device__ __forceinline__ float wave_reduce_sum(float v) {
    #pragma unroll
    for (int off = 16; off > 0; off >>= 1) v += __shfl_xor(v, off, 32);
    return v;
}

// Kernel A: per-position inverse norm + per-block partial channel sums of fn.
// grid = BATCH*16 blocks of 256 threads; block (b,chunk) owns 256 consecutive n.
__global__ void k_norm_partials(const float* __restrict__ x, float* __restrict__ ws) {
    __shared__ float ldsS[8 * CH]; // per-wave, per-channel sums
    const int b     = blockIdx.x >> 4;
    const int chunk = blockIdx.x & 15;
    const int tid   = threadIdx.x;
    const int n     = chunk * 256 + tid;
    const int lane  = tid & 31;
    const int wid   = tid >> 5;

    // pass 1: sum of squares over channels (coalesced row reads)
    float ss = 0.f;
    #pragma unroll
    for (int c = 0; c < CH; ++c) {
        float v = x[(size_t)(b * CH + c) * NPOS + n];
        ss += v * v;
    }
    float inv = 1.0f / fmaxf(sqrtf(ss), EPSN);
    ws[OFF_INV + b * NPOS + n] = inv;

    // pass 2: partial channel sums of normalized features (L2-resident re-read)
    #pragma unroll
    for (int c = 0; c < CH; ++c) {
        float v = x[(size_t)(b * CH + c) * NPOS + n] * inv;
        float wsum = wave_reduce_sum(v);
        if (lane == 0) ldsS[wid * CH + c] = wsum;
    }
    __syncthreads();
    if (tid < CH) {
        float acc = 0.f;
        #pragma unroll
        for (int w = 0; w < 8; ++w) acc += ldsS[w * CH + tid];
        ws[OFF_PART + (size_t)blockIdx.x * CH + tid] = acc;
    }
}

// Kernel B: reduce 16 chunk partials -> s[b,c]. One block, 512 threads.
__global__ void k_reduce_s(float* __restrict__ ws) {
    const int t = threadIdx.x;           // t = b*CH + c
    const int b = t >> 5;
    const int c = t & 31;
    float acc = 0.f;
    #pragma unroll
    for (int ch = 0; ch < 16; ++ch)
        acc += ws[OFF_PART + (size_t)(b * 16 + ch) * CH + c];
    ws[OFF_S + t] = acc;                 // 1/N scale dropped: top-k scale-invariant
}

// Kernel C: scores via fp32 WMMA. Each wave -> 16 positions.
// score[n] = sum_c fn[c,n] * s[c], accumulated as 8 x (16x16x4) WMMAs with the
// B matrix broadcast across columns (every result column equals the score).
__global__ void k_score_wmma(const float* __restrict__ x, float* __restrict__ ws) {
    const int wave_g = blockIdx.x * 8 + (threadIdx.x >> 5);  // 4096 waves total
    const int b    = wave_g >> 8;          // NPOS/16 = 256 tiles per batch
    const int tile = wave_g & 255;
    const int n0   = tile << 4;
    const int lane = threadIdx.x & 31;
    const int m    = lane & 15;            // matrix row within tile
    const int hi   = lane >> 4;            // lane group selects K pair {0,1} vs {2,3}

    const float* xb  = x + (size_t)b * CH * NPOS;
    const float* inv = ws + OFF_INV + (size_t)b * NPOS;
    const float* sC  = ws + OFF_S + b * CH;
    const float  invm = inv[n0 + m];

    v8f acc = {};
    #pragma unroll
    for (int k0 = 0; k0 < CH; k0 += 4) {
        const int ka = k0 + hi * 2;
        v2f a, bb;
        a.x  = xb[(size_t)(ka + 0) * NPOS + n0 + m] * invm;   // A: K=ka
        a.y  = xb[(size_t)(ka + 1) * NPOS + n0 + m] * invm;   // A: K=ka+1
        bb.x = sC[ka + 0];                                    // B row ka, all cols
        bb.y = sC[ka + 1];                                    // B row ka+1, all cols
        acc = __builtin_amdgcn_wmma_f32_16x16x4_f32(
            false, a, false, bb, (short)0, acc, false, false);
    }
    // C/D layout: VGPR r, lanes0-15 -> M=r; lanes16-31 -> M=r+8. Columns identical.
    if (m == 0) {
        float* sc = ws + OFF_SCORE + (size_t)b * NPOS + n0 + hi * 8;
        #pragma unroll
        for (int r = 0; r < 8; ++r) sc[r] = acc[r];
    }
}

// Kernel D: per-batch exact top-k mask. One block (1024 thr) per batch.
__global__ void k_topk_mask(float* __restrict__ ws) {
    __shared__ float s_sort[NPOS];   // 16 KB
    __shared__ int   scanE[1024];
    __shared__ int   scanG[1024];
    const int b   = blockIdx.x;
    const int tid = threadIdx.x;
    const float* score = ws + OFF_SCORE + (size_t)b * NPOS;
    float*       mask  = ws + OFF_MASK  + (size_t)b * NPOS;

    #pragma unroll
    for (int j = 0; j < 4; ++j) s_sort[tid + 1024 * j] = score[tid + 1024 * j];

    // bitonic sort, descending
    for (unsigned size = 2; size <= NPOS; size <<= 1) {
        for (unsigned stride = size >> 1; stride > 0; stride >>= 1) {
            __syncthreads();
            #pragma unroll
            for (int rep = 0; rep < 2; ++rep) {
                unsigned i = tid + rep * 1024;              // pair id in [0,2048)
                unsigned idx = (i / stride) * (stride * 2) + (i % stride);
                unsigned par = idx + stride;
                bool asc = ((idx & size) == 0);
                float a = s_sort[idx], c = s_sort[par];
                if ((a < c) == asc) { s_sort[idx] = c; s_sort[par] = a; }
            }
        }
    }
    __syncthreads();
    const float T = s_sort[KSEL - 1];   // k-th largest

    // per-thread counts over contiguous indices [4t, 4t+4)
    float v[4]; int cntG = 0, cntE = 0;
    #pragma unroll
    for (int j = 0; j < 4; ++j) {
        v[j] = score[tid * 4 + j];
        cntG += (v[j] > T);
        cntE += (v[j] == T);
    }
    scanE[tid] = cntE; scanG[tid] = cntG;
    for (int off = 1; off < 1024; off <<= 1) {
        __syncthreads();
        int e = scanE[tid], g = scanG[tid];
        int ea = 0, ga = 0;
        if (tid >= off) { ea = scanE[tid - off]; ga = scanG[tid - off]; }
        __syncthreads();
        scanE[tid] = e + ea; scanG[tid] = g + ga;
    }
    __syncthreads();
    const int G       = scanG[1023];
    const int need_eq = KSEL - G;                 // ties at T to keep (lowest index first)
    int eqRank = scanE[tid] - cntE;               // exclusive prefix of equals
    #pragma unroll
    for (int j = 0; j < 4; ++j) {
        float mval = 0.f;
        if (v[j] > T) mval = 1.f;
        else if (v[j] == T) { if (eqRank < need_eq) mval = 1.f; ++eqRank; }
        mask[tid * 4 + j] = mval;
    }
}

// Kernel E: out = x * mask (broadcast over channels), float4 vectorized.
__global__ void k_apply_mask(const float* __restrict__ x, const float* __restrict__ ws,
                             float* __restrict__ out) {
    const size_t q = (size_t)blockIdx.x * blockDim.x + threadIdx.x; // float4 index
    const size_t base = q * 4;                                      // total 2M floats
    const int b = (int)(base / (CH * NPOS));
    const int n = (int)(base % NPOS);
    const float4 xv = ((const float4*)x)[q];
    const float4 mv = ((const float4*)(ws + OFF_MASK + (size_t)b * NPOS + n))[0];
    float4 ov;
    ov.x = xv.x * mv.x; ov.y = xv.y * mv.y; ov.z = xv.z * mv.z; ov.w = xv.w * mv.w;
    ((float4*)out)[q] = ov;
}

extern "C" void kernel_launch(void* const* d_in, const int* in_sizes, int n_in,
                              void* d_out, int out_size, void* d_ws, size_t ws_size,
                              hipStream_t stream) {
    (void)in_sizes; (void)n_in; (void)out_size; (void)ws_size;
    const float* x  = (const float*)d_in[0];
    float*       ws = (float*)d_ws;
    float*       out = (float*)d_out;

    k_norm_partials<<<BATCH * 16, 256, 0, stream>>>(x, ws);
    k_reduce_s<<<1, BATCH * CH, 0, stream>>>(ws);
    k_score_wmma<<<(BATCH * NPOS / 16) / 8, 256, 0, stream>>>(x, ws);
    k_topk_mask<<<BATCH, 1024, 0, stream>>>(ws);
    k_apply_mask<<<(BATCH * CH * NPOS / 4) / 256, 256, 0, stream>>>(x, ws, out);
}